// node_prompt_layer_feature_cat_21534966022314
// MI455X (gfx1250) — compile-verified
//
#include <hip/hip_runtime.h>

typedef __attribute__((ext_vector_type(2))) float v2f;
typedef __attribute__((ext_vector_type(8))) float v8f;

#define N_NODES  100000
#define N_EDGES  400000
#define D_EMB    128
#define D_PROMPT 64
#define D_OUT    192   // 128 + 64

// ---------------------------------------------------------------------------
// Pass 1: zero out[:, 0:128] (scatter-accumulated) and deg[] in workspace.
// The prompt columns out[:, 128:192] are fully overwritten by pass 3, so we
// skip zeroing them to save ~25 MB of store traffic.
// ---------------------------------------------------------------------------
__global__ void zero_kernel(float* __restrict__ out, float* __restrict__ deg) {
    long tid = (long)blockIdx.x * blockDim.x + threadIdx.x;
    const long emb_elems = (long)N_NODES * D_EMB;
    if (tid < emb_elems) {
        long n = tid >> 7;           // / 128
        long c = tid & (D_EMB - 1);  // % 128
        out[n * D_OUT + c] = 0.0f;
    } else if (tid < emb_elems + N_NODES) {
        deg[tid - emb_elems] = 0.0f;
    }
}

__device__ inline void atomic_fadd(float* p, float v) {
    __hip_atomic_fetch_add(p, v, __ATOMIC_RELAXED, __HIP_MEMORY_SCOPE_AGENT);
}

// ---------------------------------------------------------------------------
// Pass 2: one wave32 per edge. Lane L loads emb[src][4L..4L+3] as a single
// b128 (512B coalesced per row) and atomically accumulates into out[dst].
// Lane 0 counts the in-degree.
// ---------------------------------------------------------------------------
__global__ void scatter_kernel(const float* __restrict__ emb,
                               const int*   __restrict__ src,
                               const int*   __restrict__ dst,
                               float* __restrict__ out,
                               float* __restrict__ deg) {
    const int lane = threadIdx.x & 31;
    const int wave = threadIdx.x >> 5;
    const long e = (long)blockIdx.x * (blockDim.x >> 5) + wave;
    if (e >= N_EDGES) return;

    const int s = src[e];
    const int d = dst[e];

    const float4* row = (const float4*)(emb + (long)s * D_EMB);
    float4 v = row[lane];

    float* o = out + (long)d * D_OUT + lane * 4;
    atomic_fadd(o + 0, v.x);
    atomic_fadd(o + 1, v.y);
    atomic_fadd(o + 2, v.z);
    atomic_fadd(o + 3, v.w);

    if (lane == 0) atomic_fadd(deg + d, 1.0f);
}

// ---------------------------------------------------------------------------
// Pass 3: out[n, 128+j] = deg[n] * weight[j] — rank-1 outer product done on
// the matrix pipe. One wave per 16-node tile; 4x V_WMMA_F32_16X16X4_F32 per
// tile (one per 16-wide weight chunk).
//
// A (16x4 f32): lanes 0-15 hold M=0..15 with VGPR0=K0, VGPR1=K1; lanes 16-31
//   hold K2/K3. We place deg in K=0 (lanes 0-15, a.x) and zero elsewhere.
// B (4x16 f32): by symmetry, VGPR0 lanes 0-15 hold K=0, N=lane. Weight chunk
//   goes in b.x of lanes 0-15; everything else zero.
// D (16x16 f32): VGPR r -> M=r (lanes 0-15) / M=8+r (lanes 16-31), N=lane%16.
// ---------------------------------------------------------------------------
__global__ void prompt_wmma_kernel(const float* __restrict__ deg,
                                   const float* __restrict__ weight,
                                   float* __restrict__ out) {
    const int lane = threadIdx.x & 31;
    const int wave = threadIdx.x >> 5;
    const int tile = blockIdx.x * (blockDim.x >> 5) + wave;  // uniform per wave
    if (tile >= N_NODES / 16) return;                         // wave-uniform exit

    const int lo   = (lane < 16);
    const int l15  = lane & 15;

    v2f a;
    a.x = lo ? deg[tile * 16 + l15] : 0.0f;  // A[m][k=0] = deg
    a.y = 0.0f;

    #pragma unroll
    for (int j = 0; j < 4; ++j) {
        v2f b;
        b.x = lo ? weight[j * 16 + l15] : 0.0f;  // B[k=0][n] = weight chunk
        b.y = 0.0f;

        v8f c = {};
        // 8 args: (neg_a, A, neg_b, B, c_mod, C, reuse_a, reuse_b)
        c = __builtin_amdgcn_wmma_f32_16x16x4_f32(
                false, a, false, b, (short)0, c, false, false);

        const int col = D_EMB + j * 16 + l15;
        #pragma unroll
        for (int r = 0; r < 8; ++r) {
            const int m   = lo ? r : (8 + r);
            const long row = (long)(tile * 16 + m);
            out[row * D_OUT + col] = c[r];
        }
    }
}

extern "C" void kernel_launch(void* const* d_in, const int* in_sizes, int n_in,
                              void* d_out, int out_size, void* d_ws, size_t ws_size,
                              hipStream_t stream) {
    const float* emb    = (const float*)d_in[0];  // [100000, 128] f32
    const float* weight = (const float*)d_in[1];  // [1, 64] f32
    const int*   src    = (const int*)d_in[2];    // [400000] int
    const int*   dst    = (const int*)d_in[3];    // [400000] int

    float* out = (float*)d_out;                   // [100000, 192] f32
    float* deg = (float*)d_ws;                    // [100000] f32 scratch

    // Pass 1: zero accumulation targets.
    {
        long n = (long)N_NODES * D_EMB + N_NODES;
        int blocks = (int)((n + 255) / 256);
        zero_kernel<<<blocks, 256, 0, stream>>>(out, deg);
    }

    // Pass 2: edge scatter (one wave32 per edge, 8 waves per 256-thread block).
    {
        int waves_per_block = 256 / 32;
        int blocks = (N_EDGES + waves_per_block - 1) / waves_per_block;
        scatter_kernel<<<blocks, 256, 0, stream>>>(emb, src, dst, out, deg);
    }

    // Pass 3: prompt outer product via WMMA (one wave per 16-node tile).
    {
        int tiles = N_NODES / 16;  // 6250, exact
        int waves_per_block = 256 / 32;
        int blocks = (tiles + waves_per_block - 1) / waves_per_block;
        prompt_wmma_kernel<<<blocks, 256, 0, stream>>>(deg, weight, out);
    }
}